// PolygonLoss_21732534518045
// MI455X (gfx1250) — compile-verified
//
#include <hip/hip_runtime.h>
#include <stdint.h>

// Problem constants from the reference
#define B_ 64
#define M_ 64
#define P_ 128
#define V_ 32
#define EPSF 1e-12f
#define FLTBIG 3.4e38f

// ---- CDNA5 async global->LDS helpers (inline asm; ASYNCcnt path) ----------

// Low 32 bits of a generic pointer to a __shared__ object == wave-relative LDS
// byte offset (HW adds LDS_BASE for async-to-LDS ops).
__device__ __forceinline__ uint32_t lds_byte_off(const void* p) {
  return (uint32_t)(uintptr_t)p;
}

// Each active lane copies 16 bytes from the global address gaddr to the LDS offset lds_addr.
// GV addressing mode (64-bit vaddr, saddr = off). Tracked with ASYNCcnt.
__device__ __forceinline__ void async_load_to_lds_b128(uint32_t lds_addr,
                                                       const void* gaddr) {
  asm volatile("global_load_async_to_lds_b128 %0, %1, off"
               :: "v"(lds_addr), "v"(gaddr)
               : "memory");
}

__device__ __forceinline__ void wait_async_zero() {
  asm volatile("s_wait_asynccnt 0x0" ::: "memory");
}

// ---- Kernel 1: each block handles 4 polygons x 64 points, one pair/thread --

__global__ __launch_bounds__(256) void polygon_dist_partial(
    const float* __restrict__ last_point,      // [B, M, 2]
    const float* __restrict__ polygon_coords,  // [B, P, V, 2]
    float* __restrict__ block_min)             // [32]
{
  __shared__ float2 sPoly[4 * V_];  // 4 polygons x 32 verts, 1 KB
  __shared__ float2 sPts[M_];       // 64 points, 512 B
  __shared__ float  sRed[8];        // cross-wave reduction (8 waves)

  const int tid = threadIdx.x;

  // Only batch B-1 = 63 contributes to the output.
  const float* lp63 = last_point     + (size_t)(B_ - 1) * M_ * 2;
  const float* pg63 = polygon_coords + (size_t)(B_ - 1) * P_ * V_ * 2;

  // ---- Stage data into LDS with the async copy engine --------------------
  // 64 chunks of 16B for the 4 polygons (waves 0-1), 32 chunks for the
  // 64 points (wave 2). Waves 3-7 issue nothing (ASYNCcnt stays 0).
  if (tid < 64) {
    const int q   = tid >> 4;   // polygon 0..3 within block
    const int sub = tid & 15;   // 16B chunk within polygon (256 B each)
    const float* src = pg63 + ((size_t)(blockIdx.x * 4 + q) * V_ * 2) + sub * 4;
    async_load_to_lds_b128(lds_byte_off(&sPoly[0]) + tid * 16, src);
  } else if (tid < 96) {
    const int c = tid - 64;     // 16B chunk of the point array
    const float* src = lp63 + c * 4;
    async_load_to_lds_b128(lds_byte_off(&sPts[0]) + c * 16, src);
  }
  wait_async_zero();   // issuing waves drain their own ASYNCcnt
  __syncthreads();     // publish LDS to all waves

  // ---- One (point m, polygon q) pair per thread --------------------------
  // q uniform per wave -> polygon vertex reads are LDS broadcasts.
  const int q = tid >> 6;     // 0..3
  const int m = tid & 63;     // 0..63

  const float2 pt = sPts[m];
  const float px = pt.x, py = pt.y;

  const float2* vp = &sPoly[q * V_];

  float mind2 = FLTBIG;
  int   cross = 0;
  float2 a = vp[0];
  for (int k = 0; k < V_; ++k) {
    const float2 b = vp[(k + 1) & (V_ - 1)];
    // segment distance (matches reference: clipped projection)
    const float abx = b.x - a.x, aby = b.y - a.y;
    const float apx = px - a.x,  apy = py - a.y;
    float t = (apx * abx + apy * aby) / (abx * abx + aby * aby + EPSF);
    t = fminf(fmaxf(t, 0.0f), 1.0f);
    const float dx = apx - t * abx, dy = apy - t * aby;
    const float d2 = fmaxf(dx * dx + dy * dy, EPSF);
    mind2 = fminf(mind2, d2);
    // ray-crossing parity test (matches reference formula incl. EPS)
    const bool straddle = (a.y > py) != (b.y > py);
    const float xi = a.x + (b.x - a.x) * (py - a.y) / (b.y - a.y + EPSF);
    cross += (straddle && (px < xi)) ? 1 : 0;
    a = b;
  }
  // min of sqrt == sqrt of min (monotone), so defer the sqrt
  float dist = (cross & 1) ? 0.0f : sqrtf(mind2);

  // ---- Block min reduction (wave32 shuffles, then LDS across 8 waves) ----
  for (int off = 16; off > 0; off >>= 1)
    dist = fminf(dist, __shfl_xor(dist, off, 32));
  const int wave = tid >> 5;
  const int lane = tid & 31;
  if (lane == 0) sRed[wave] = dist;
  __syncthreads();
  if (wave == 0) {
    float v = (lane < 8) ? sRed[lane] : FLTBIG;
    for (int off = 4; off > 0; off >>= 1)
      v = fminf(v, __shfl_xor(v, off, 32));
    if (lane == 0) block_min[blockIdx.x] = v;
  }
}

// ---- Kernel 2: single wave folds the 32 partial minima ---------------------

__global__ __launch_bounds__(32) void polygon_dist_final(
    const float* __restrict__ block_min, float* __restrict__ out)
{
  const int lane = threadIdx.x;
  float v = block_min[lane];
  for (int off = 16; off > 0; off >>= 1)
    v = fminf(v, __shfl_xor(v, off, 32));
  if (lane == 0) out[0] = 0.5f * v;  // POLYGON_LOSS_WEIGHT
}

// ---- Entry -----------------------------------------------------------------

extern "C" void kernel_launch(void* const* d_in, const int* in_sizes, int n_in,
                              void* d_out, int out_size, void* d_ws, size_t ws_size,
                              hipStream_t stream) {
  (void)in_sizes; (void)n_in; (void)out_size; (void)ws_size;
  const float* last_point     = (const float*)d_in[0];  // [64,64,2] f32
  const float* polygon_coords = (const float*)d_in[1];  // [64,128,32,2] f32
  float* out = (float*)d_out;                           // 1 x f32
  float* ws  = (float*)d_ws;                            // 32 x f32 partials

  polygon_dist_partial<<<32, 256, 0, stream>>>(last_point, polygon_coords, ws);
  polygon_dist_final<<<1, 32, 0, stream>>>(ws, out);
}